// EdgeAwareLoss_28200755265882
// MI455X (gfx1250) — compile-verified
//
#include <hip/hip_runtime.h>

// ---------------------------------------------------------------------------
// Sobel edge-magnitude L1 loss, gfx1250 (MI455X). HBM-bound (2x128MiB fp32 in,
// scalar out; ~11.5us floor at 23.3 TB/s). CDNA5 path: B128 async global->LDS
// tile copies + s_wait_asynccnt, LDS stencil with register sliding window,
// wave32 shuffle reduction, f32 global atomic accumulation.
// ---------------------------------------------------------------------------

#define TH     32            // output rows per workgroup (6.25% halo overhead)
#define HROWS  (TH + 2)      // tile rows incl. halo = 34
#define SW     264           // LDS row stride (floats); data cols 4..259
#define DC     4             // first data column in padded coords
#define IMG_W  256
#define IMG_H  256

#if defined(__gfx1250__) && __has_builtin(__builtin_amdgcn_global_load_async_to_lds_b128)
#define ASYNC_B128 1
#else
#define ASYNC_B128 0
#endif
#if defined(__gfx1250__) && __has_builtin(__builtin_amdgcn_global_load_async_to_lds_b32)
#define ASYNC_B32 1
#else
#define ASYNC_B32 0
#endif

typedef int v4i __attribute__((vector_size(16)));
typedef __attribute__((address_space(1))) int* as1_i32p;
typedef __attribute__((address_space(3))) int* as3_i32p;
typedef __attribute__((address_space(1))) v4i* as1_v4ip;
typedef __attribute__((address_space(3))) v4i* as3_v4ip;

// Copy 16 bytes (4 floats) global -> LDS. Both addresses 16B-aligned.
__device__ __forceinline__ void tile_copy_f32x4(const float* gsrc, float* ldst) {
#if ASYNC_B128
    __builtin_amdgcn_global_load_async_to_lds_b128(
        (as1_v4ip)(unsigned long long)gsrc,
        (as3_v4ip)(unsigned int)(unsigned long long)ldst,
        /*imm offset*/0, /*cpol*/0);
#elif ASYNC_B32
#pragma unroll
    for (int i = 0; i < 4; ++i)
        __builtin_amdgcn_global_load_async_to_lds_b32(
            (as1_i32p)(unsigned long long)(gsrc + i),
            (as3_i32p)(unsigned int)(unsigned long long)(ldst + i),
            0, 0);
#else
    *(float4*)ldst = *(const float4*)gsrc;
#endif
}

__device__ __forceinline__ void wait_async_lds0() {
#if ASYNC_B128 || ASYNC_B32
#if __has_builtin(__builtin_amdgcn_s_wait_asynccnt)
    __builtin_amdgcn_s_wait_asynccnt(0);
#else
    asm volatile("s_wait_asynccnt 0" ::: "memory");
#endif
#endif
}

__global__ void edge_loss_zero_kernel(float* out) {
    if (threadIdx.x == 0 && blockIdx.x == 0) out[0] = 0.0f;
}

__global__ __launch_bounds__(256)
void edge_loss_sobel_kernel(const float* __restrict__ pred,
                            const float* __restrict__ targ,
                            float* __restrict__ out,
                            float inv_total) {
    __shared__ float sp[HROWS * SW];
    __shared__ float st[HROWS * SW];
    __shared__ float wave_sums[8];

    const int t  = threadIdx.x;              // 0..255
    const int r0 = blockIdx.x * TH;          // first output row of this band
    const long long slice = blockIdx.y;      // 0..511 (B*D slices)
    const float* gp = pred + slice * (long long)(IMG_H * IMG_W);
    const float* gt = targ + slice * (long long)(IMG_H * IMG_W);

    // Zero halo columns (padded col 3 and col 260) for all tile rows.
    if (t < HROWS) {
        sp[t * SW + (DC - 1)]     = 0.0f;
        sp[t * SW + (DC + IMG_W)] = 0.0f;
        st[t * SW + (DC - 1)]     = 0.0f;
        st[t * SW + (DC + IMG_W)] = 0.0f;
    }

    // Fill tile rows r0-1 .. r0+TH. Each step: 256 threads = 4 rows x 64 lanes
    // of float4 (B128 async, 16B-aligned LDS dest). Rows outside image -> 0.
    {
        const int sub    = t >> 6;           // 0..3: row within the step
        const int lane64 = t & 63;           // 0..63: float4 index in the row
        const float zero4[4] = {0.f, 0.f, 0.f, 0.f};
#pragma unroll
        for (int s = 0; s < (HROWS + 3) / 4; ++s) {
            const int rr = s * 4 + sub;      // tile row (uniform per wave)
            if (rr < HROWS) {
                const int gr = r0 + rr - 1;  // global row
                const int li = rr * SW + DC + lane64 * 4;
                if (gr >= 0 && gr < IMG_H) {
                    const long long go = (long long)gr * IMG_W + lane64 * 4;
                    tile_copy_f32x4(gp + go, &sp[li]);
                    tile_copy_f32x4(gt + go, &st[li]);
                } else {
                    *(float4*)&sp[li] = *(const float4*)zero4;
                    *(float4*)&st[li] = *(const float4*)zero4;
                }
            }
        }
    }

    wait_async_lds0();
    __syncthreads();

    // Sliding 3x3 window down one column (c = t+DC in padded coords).
    const int c = t + DC;
    float p00 = sp[0 * SW + c - 1], p01 = sp[0 * SW + c], p02 = sp[0 * SW + c + 1];
    float p10 = sp[1 * SW + c - 1], p11 = sp[1 * SW + c], p12 = sp[1 * SW + c + 1];
    float q00 = st[0 * SW + c - 1], q01 = st[0 * SW + c], q02 = st[0 * SW + c + 1];
    float q10 = st[1 * SW + c - 1], q11 = st[1 * SW + c], q12 = st[1 * SW + c + 1];

    float acc = 0.0f;
#pragma unroll
    for (int ry = 0; ry < TH; ++ry) {
        const int rb = (ry + 2) * SW + c;
        const float p20 = sp[rb - 1], p21 = sp[rb], p22 = sp[rb + 1];
        const float q20 = st[rb - 1], q21 = st[rb], q22 = st[rb + 1];

        // XLA conv is cross-correlation:
        // ex = (right col . [1,2,1]) - (left col . [1,2,1])
        // ey = (bottom row . [1,2,1]) - (top row . [1,2,1])
        const float pex = (p02 + 2.0f * p12 + p22) - (p00 + 2.0f * p10 + p20);
        const float pey = (p20 + 2.0f * p21 + p22) - (p00 + 2.0f * p01 + p02);
        const float qex = (q02 + 2.0f * q12 + q22) - (q00 + 2.0f * q10 + q20);
        const float qey = (q20 + 2.0f * q21 + q22) - (q00 + 2.0f * q01 + q02);

        const float pm = sqrtf(fmaf(pex, pex, fmaf(pey, pey, 1e-8f)));
        const float qm = sqrtf(fmaf(qex, qex, fmaf(qey, qey, 1e-8f)));
        acc += fabsf(pm - qm);

        p00 = p10; p01 = p11; p02 = p12;
        p10 = p20; p11 = p21; p12 = p22;
        q00 = q10; q01 = q11; q02 = q12;
        q10 = q20; q11 = q21; q12 = q22;
    }

    // wave32 reduction
#pragma unroll
    for (int off = 16; off > 0; off >>= 1)
        acc += __shfl_down(acc, off, 32);

    const int lane = t & 31;
    const int wid  = t >> 5;
    if (lane == 0) wave_sums[wid] = acc;
    __syncthreads();

    if (wid == 0) {
        float v = (lane < 8) ? wave_sums[lane] : 0.0f;
#pragma unroll
        for (int off = 4; off > 0; off >>= 1)
            v += __shfl_down(v, off, 32);
        if (lane == 0) atomicAdd(out, v * inv_total);
    }
}

extern "C" void kernel_launch(void* const* d_in, const int* in_sizes, int n_in,
                              void* d_out, int out_size, void* d_ws, size_t ws_size,
                              hipStream_t stream) {
    const float* pred = (const float*)d_in[0];
    const float* targ = (const float*)d_in[1];
    float* out = (float*)d_out;

    const long long total = (long long)in_sizes[0];           // B*C*D*H*W
    const int slices = (int)(total / (IMG_H * IMG_W));        // B*D = 512
    const float inv_total = 1.0f / (float)total;

    edge_loss_zero_kernel<<<1, 32, 0, stream>>>(out);

    dim3 grid(IMG_H / TH, slices);
    edge_loss_sobel_kernel<<<grid, 256, 0, stream>>>(pred, targ, out, inv_total);
}